// PCEN_90280212562198
// MI455X (gfx1250) — compile-verified
//
#include <hip/hip_runtime.h>

#define FLOORV 1e-12f

constexpr int kB = 64, kC = 80, kT = 8000;
constexpr int KPL    = 16;                       // floats per lane per tile
constexpr int TILE   = 32 * KPL;                 // 512 floats per tile
constexpr int NTILES = (kT + TILE - 1) / TILE;   // 16 tiles (last partial)
constexpr int TAILF  = kT - (NTILES - 1) * TILE; // 320 floats in last tile
constexpr int TAIL_LANES = TAILF / KPL;          // 20 full lanes (8000 % 16 == 0)
constexpr int PADQ   = 5;                        // float4 slots per lane (4 data + 1 pad)
constexpr int WPB    = 8;                        // waves per block
constexpr int QPW    = 32 * PADQ;                // float4 slots per buffer per wave

__device__ __forceinline__ float fast_log2(float v) { return __builtin_amdgcn_logf(v); }
__device__ __forceinline__ float fast_exp2(float v) { return __builtin_amdgcn_exp2f(v); }

// Issue 4 async b128 global->LDS copies for one 512-float tile of one row.
// Each active lane moves its own 64B contiguous segment into a padded LDS slot.
__device__ __forceinline__ void issue_tile_async(const float* __restrict__ xrow,
                                                 int t, int wave, int buf, int lane) {
    const int valid = (t == NTILES - 1) ? TAIL_LANES : 32;
    if (lane < valid) {
        const char* g = (const char*)(xrow + (size_t)t * TILE + lane * KPL);
        unsigned l0 = (unsigned)(((wave * 2 + buf) * QPW + lane * PADQ) * 16);
#pragma unroll
        for (int q = 0; q < 4; ++q) {
            asm volatile("global_load_async_to_lds_b128 %0, %1, off"
                         :
                         : "v"(l0 + (unsigned)(q * 16)),
                           "v"((unsigned long long)(uintptr_t)(g + q * 16))
                         : "memory");
        }
    }
}

__global__ void __launch_bounds__(WPB * 32)
PCEN_90280212562198_kernel(const float* __restrict__ x,
                           const float* __restrict__ alpha,
                           const float* __restrict__ delta,
                           const float* __restrict__ root,
                           const float* __restrict__ ema_w,
                           float* __restrict__ out) {
    // Single shared array -> starts at LDS byte offset 0 (matches asm LDS addresses).
    __shared__ float4 smem[WPB * 2 * QPW];

    const int lane = threadIdx.x & 31;
    const int wave = threadIdx.x >> 5;
    const int row  = blockIdx.x * WPB + wave;
    if (row >= kB * kC) return;

    const int ch = row % kC;
    const float w_  = fminf(fmaxf(ema_w[ch], 0.0f), 1.0f);
    const float om  = 1.0f - w_;
    const float a_  = fminf(alpha[ch], 1.0f);
    const float r_  = fmaxf(root[ch], 1.0f);
    const float oor = 1.0f / r_;
    const float d_  = delta[ch];
    const float droot = fast_exp2(oor * fast_log2(d_));   // d^(1/r)

    const float om2 = om * om, om4 = om2 * om2, om8 = om4 * om4, om16 = om8 * om8;

    const float* xrow = x + (size_t)row * kT;
    float*       orow = out + (size_t)row * kT;

    // Recurrence m[t] = w*x[t] + (1-w)*m[t-1] with m[-1] := x[0] reproduces
    // the reference scan exactly (m[0] = w*x0 + (1-w)*x0).
    float carry = xrow[0];

    issue_tile_async(xrow, 0, wave, 0, lane);

    for (int t = 0; t < NTILES; ++t) {
        const int buf   = t & 1;
        const int valid = (t == NTILES - 1) ? TAIL_LANES : 32;

        if (t + 1 < NTILES) {
            issue_tile_async(xrow, t + 1, wave, buf ^ 1, lane);
            asm volatile("s_wait_asynccnt 4" ::: "memory");   // tile t complete, t+1 in flight
        } else {
            asm volatile("s_wait_asynccnt 0" ::: "memory");
        }

        // Read my contiguous 16-element segment from padded LDS (2-way b128, optimal).
        const float4* src = &smem[(wave * 2 + buf) * QPW + lane * PADQ];
        float4 q0 = src[0], q1 = src[1], q2 = src[2], q3 = src[3];
        float e[KPL] = { q0.x, q0.y, q0.z, q0.w,  q1.x, q1.y, q1.z, q1.w,
                         q2.x, q2.y, q2.z, q2.w,  q3.x, q3.y, q3.z, q3.w };

        // Pass 1: local EMA with zero carry -> affine map (A, B) for this lane's segment.
        float p = 0.0f;
#pragma unroll
        for (int j = 0; j < KPL; ++j) p = fmaf(om, p, w_ * e[j]);
        float Av = om16;   // A = (1-w)^16
        float Bv = p;      // B = partial_15

        // Wave32 inclusive scan of affine composition (A,B) o (A',B') = (A*A', A*B'+B).
#pragma unroll
        for (int dlt = 1; dlt < 32; dlt <<= 1) {
            float Au = __shfl_up(Av, dlt, 32);
            float Bu = __shfl_up(Bv, dlt, 32);
            if (lane >= dlt) { Bv = fmaf(Av, Bu, Bv); Av *= Au; }
        }
        const float inc        = fmaf(Av, carry, Bv);      // F_l(carry)
        float       Mexc       = __shfl_up(inc, 1, 32);    // exclusive carry for this lane
        if (lane == 0) Mexc = carry;
        const float carry_next = __shfl(inc, 31, 32);      // carry into next tile

        // Pass 2: rerun recurrence with true carry; fuse pointwise PCEN transform.
        float mm = Mexc;
        float o[KPL];
#pragma unroll
        for (int j = 0; j < KPL; ++j) {
            mm = fmaf(om, mm, w_ * e[j]);                  // m at this position
            const float lg  = fast_log2(FLOORV + mm);
            const float den = fast_exp2(-a_ * lg);         // (m+eps)^-a
            const float v   = fmaf(e[j], den, d_);         // x*(m+eps)^-a + d
            o[j] = fast_exp2(oor * fast_log2(v)) - droot;
        }

        if (lane < valid) {
            float4* dst = (float4*)(orow + (size_t)t * TILE + lane * KPL);
            dst[0] = make_float4(o[0],  o[1],  o[2],  o[3]);
            dst[1] = make_float4(o[4],  o[5],  o[6],  o[7]);
            dst[2] = make_float4(o[8],  o[9],  o[10], o[11]);
            dst[3] = make_float4(o[12], o[13], o[14], o[15]);
        }
        carry = carry_next;
    }
}

// x_length passes through unchanged (bit-copy int32 into the output tail).
__global__ void PCEN_copy_len(const int* __restrict__ xl, int* __restrict__ dst, int n) {
    int i = blockIdx.x * blockDim.x + threadIdx.x;
    if (i < n) dst[i] = xl[i];
}

extern "C" void kernel_launch(void* const* d_in, const int* in_sizes, int n_in,
                              void* d_out, int out_size, void* d_ws, size_t ws_size,
                              hipStream_t stream) {
    (void)in_sizes; (void)n_in; (void)d_ws; (void)ws_size; (void)out_size;
    const float* x     = (const float*)d_in[0];
    const float* alpha = (const float*)d_in[1];
    const float* delta = (const float*)d_in[2];
    const float* root  = (const float*)d_in[3];
    const float* ema_w = (const float*)d_in[4];
    const int*   xlen  = (const int*)d_in[5];
    float* out = (float*)d_out;

    const int rows = kB * kC;                 // 5120 rows, one wave each
    dim3 grid(rows / WPB), block(WPB * 32);   // 640 blocks x 256 threads
    PCEN_90280212562198_kernel<<<grid, block, 0, stream>>>(x, alpha, delta, root, ema_w, out);
    PCEN_copy_len<<<1, 64, 0, stream>>>(xlen, (int*)(out + (size_t)kB * kC * kT), kB);
}